// SAGNN_79259326480902
// MI455X (gfx1250) — compile-verified
//
#include <hip/hip_runtime.h>

// ---------------------------------------------------------------------------
// SAGNN on gfx1250: fp32 WMMA (V_WMMA_F32_16X16X4_F32) for all GEMMs,
// async global->LDS (GLOBAL_LOAD_ASYNC_TO_LDS_B128) for weight staging,
// uint-bit atomicMax for the relu'd segment_max scatter.
// ---------------------------------------------------------------------------

typedef float v2f __attribute__((ext_vector_type(2)));
typedef float v8f __attribute__((ext_vector_type(8)));

#define WMMA_F32X4(a, b, c) \
  __builtin_amdgcn_wmma_f32_16x16x4_f32(false, (a), false, (b), (short)0, (c), false, false)

// Async-copy nFloats (multiple of 1024) from global to LDS; 16B per lane per
// iteration via GLOBAL_LOAD_ASYNC_TO_LDS_B128 (tracked by ASYNCcnt).
// Caller must issue async_wait_all() + __syncthreads() before reading.
__device__ __forceinline__ void async_copy_w(const float* __restrict__ g,
                                             float* s, int nFloats) {
  for (int i = threadIdx.x * 4; i < nFloats; i += 256 * 4) {
    unsigned lds_off = (unsigned)(size_t)(s + i);            // lo32 of generic = LDS offset
    unsigned long long gaddr = (unsigned long long)(size_t)(g + i);
    asm volatile("global_load_async_to_lds_b128 %0, %1, off"
                 :: "v"(lds_off), "v"(gaddr)
                 : "memory");
  }
}

__device__ __forceinline__ void async_wait_all() {
  asm volatile("s_wait_asynccnt 0x0" ::: "memory");
}

// One K=4 step: A row fragment from xrow, B from a [K,64] weight (LDS),
// accumulating four 16x16 column tiles (cols 0..63).
__device__ __forceinline__ void wmma_k4(const float* __restrict__ xrow,
                                        const float* __restrict__ W,
                                        int kk, int g2, int ncol,
                                        v8f& c0, v8f& c1, v8f& c2, v8f& c3) {
  v2f a;
  a.x = xrow[kk + g2 + 0];
  a.y = xrow[kk + g2 + 1];
  int kr0 = (kk + g2) * 64 + ncol;   // B vgpr0: K = kk + g2 + 0
  int kr1 = kr0 + 64;                // B vgpr1: K = kk + g2 + 1
  v2f b;
  b.x = W[kr0 +  0]; b.y = W[kr1 +  0]; c0 = WMMA_F32X4(a, b, c0);
  b.x = W[kr0 + 16]; b.y = W[kr1 + 16]; c1 = WMMA_F32X4(a, b, c1);
  b.x = W[kr0 + 32]; b.y = W[kr1 + 32]; c2 = WMMA_F32X4(a, b, c2);
  b.x = W[kr0 + 48]; b.y = W[kr1 + 48]; c3 = WMMA_F32X4(a, b, c3);
}

// ps = x @ Wsrc ; pd = x @ Wdst   (each [N,64] = [N,64]x[64,64])
// One wave -> one (16-row tile, matrix) pair, 4 col tiles.
__global__ __launch_bounds__(256)
void sagnn_proj_kernel(const float* __restrict__ x,
                       const float* __restrict__ Wsrc,
                       const float* __restrict__ Wdst,
                       float* __restrict__ ps, float* __restrict__ pd,
                       int nRowTiles) {
  __shared__ alignas(16) float sW[2][64 * 64];
  async_copy_w(Wsrc, sW[0], 4096);
  async_copy_w(Wdst, sW[1], 4096);
  async_wait_all();
  __syncthreads();

  int wave = threadIdx.x >> 5;
  int lane = threadIdx.x & 31;
  int gw = blockIdx.x * 8 + wave;
  int rt = gw >> 1;
  int which = gw & 1;
  if (rt >= nRowTiles) return;

  const float* W = sW[which];
  float* out = which ? pd : ps;
  int row  = rt * 16 + (lane & 15);
  int g2   = (lane >> 4) * 2;
  int ncol = lane & 15;

  v8f c0 = {}, c1 = {}, c2 = {}, c3 = {};
  const float* xrow = x + (size_t)row * 64;
  for (int kk = 0; kk < 64; kk += 4)
    wmma_k4(xrow, W, kk, g2, ncol, c0, c1, c2, c3);

  int m0 = rt * 16 + 8 * (lane >> 4);
  for (int r = 0; r < 8; ++r) {
    float* o = out + (size_t)(m0 + r) * 64 + ncol;
    o[0]  = c0[r];
    o[16] = c1[r];
    o[32] = c2[r];
    o[48] = c3[r];
  }
}

// Zero n-buffer (segment-max identity for relu'd values is 0).
__global__ __launch_bounds__(256)
void sagnn_zero_kernel(float* __restrict__ p, int n) {
  int i = blockIdx.x * 256 + threadIdx.x;
  if (i < n) p[i] = 0.0f;
}

// Per edge: sf = relu(ps[src]*coef_src[fe]), df = relu(pd[dst]*coef_dst[se])
// atomic max into n[dst, 0:64] / n[dst, 64:128]. 64 threads per edge.
__global__ __launch_bounds__(256)
void sagnn_edge_kernel(const float* __restrict__ ps, const float* __restrict__ pd,
                       const int* __restrict__ src, const int* __restrict__ dst,
                       const int* __restrict__ fe, const int* __restrict__ se,
                       const float* __restrict__ coef_s,  // 2 floats (this layer)
                       const float* __restrict__ coef_d,  // 2 floats
                       float* __restrict__ nbuf, int E) {
  unsigned gid = blockIdx.x * 256 + threadIdx.x;
  int e = (int)(gid >> 6);
  if (e >= E) return;
  int c = (int)(gid & 63);
  int s = src[e];
  int d = dst[e];
  float cs = coef_s[fe[e]];
  float cd = coef_d[se[e]];
  float sv = ps[(size_t)s * 64 + c] * cs;
  float dv = pd[(size_t)d * 64 + c] * cd;
  sv = sv > 0.0f ? sv : 0.0f;   // ensure +0 (avoid -0 bit pattern)
  dv = dv > 0.0f ? dv : 0.0f;
  // Non-negative floats are order-isomorphic to their uint bit patterns.
  unsigned* nb = (unsigned*)nbuf;
  atomicMax(nb + (size_t)d * 128 + c,       __float_as_uint(sv));
  atomicMax(nb + (size_t)d * 128 + 64 + c,  __float_as_uint(dv));
}

// z = relu(n @ Wlin + blin + x @ Wskip + bskip)   ([N,128]x[128,64] + [N,64]x[64,64])
__global__ __launch_bounds__(256)
void sagnn_combine_kernel(const float* __restrict__ nbuf, const float* __restrict__ x,
                          const float* __restrict__ Wlin, const float* __restrict__ blin,
                          const float* __restrict__ Wskip, const float* __restrict__ bskip,
                          float* __restrict__ z, int nRowTiles) {
  __shared__ alignas(16) float sWl[128 * 64];
  __shared__ alignas(16) float sWs[64 * 64];
  __shared__ float sb[64];
  async_copy_w(Wlin, sWl, 8192);
  async_copy_w(Wskip, sWs, 4096);
  if (threadIdx.x < 64) sb[threadIdx.x] = blin[threadIdx.x] + bskip[threadIdx.x];
  async_wait_all();
  __syncthreads();

  int wave = threadIdx.x >> 5;
  int lane = threadIdx.x & 31;
  int rt = blockIdx.x * 8 + wave;
  if (rt >= nRowTiles) return;

  int row  = rt * 16 + (lane & 15);
  int g2   = (lane >> 4) * 2;
  int ncol = lane & 15;

  v8f c0 = {}, c1 = {}, c2 = {}, c3 = {};
  const float* arow = nbuf + (size_t)row * 128;
  for (int kk = 0; kk < 128; kk += 4)
    wmma_k4(arow, sWl, kk, g2, ncol, c0, c1, c2, c3);
  const float* xrow = x + (size_t)row * 64;
  for (int kk = 0; kk < 64; kk += 4)
    wmma_k4(xrow, sWs, kk, g2, ncol, c0, c1, c2, c3);

  int m0 = rt * 16 + 8 * (lane >> 4);
  for (int r = 0; r < 8; ++r) {
    float* o = z + (size_t)(m0 + r) * 64 + ncol;
    float v0 = c0[r] + sb[ncol];      o[0]  = v0 > 0.0f ? v0 : 0.0f;
    float v1 = c1[r] + sb[ncol + 16]; o[16] = v1 > 0.0f ? v1 : 0.0f;
    float v2 = c2[r] + sb[ncol + 32]; o[32] = v2 > 0.0f ? v2 : 0.0f;
    float v3 = c3[r] + sb[ncol + 48]; o[48] = v3 > 0.0f ? v3 : 0.0f;
  }
}

// x = z / ||z||_2  (row-wise, zero norm -> 1). One wave per row.
__global__ __launch_bounds__(256)
void sagnn_normalize_kernel(const float* __restrict__ z, float* __restrict__ xo, int N) {
  int wave = threadIdx.x >> 5;
  int lane = threadIdx.x & 31;
  int row = blockIdx.x * 8 + wave;
  if (row >= N) return;
  const float* zr = z + (size_t)row * 64;
  float a = zr[lane];
  float b = zr[lane + 32];
  float s = a * a + b * b;
  for (int off = 16; off > 0; off >>= 1) s += __shfl_xor(s, off, 32);
  float nrm = sqrtf(s);
  float inv = (nrm == 0.0f) ? 1.0f : 1.0f / nrm;
  float* xr = xo + (size_t)row * 64;
  xr[lane]      = a * inv;
  xr[lane + 32] = b * inv;
}

// out = x @ Wout + bout
__global__ __launch_bounds__(256)
void sagnn_out_kernel(const float* __restrict__ x, const float* __restrict__ W,
                      const float* __restrict__ bias, float* __restrict__ out,
                      int nRowTiles) {
  __shared__ alignas(16) float sW[64 * 64];
  __shared__ float sb[64];
  async_copy_w(W, sW, 4096);
  if (threadIdx.x < 64) sb[threadIdx.x] = bias[threadIdx.x];
  async_wait_all();
  __syncthreads();

  int wave = threadIdx.x >> 5;
  int lane = threadIdx.x & 31;
  int rt = blockIdx.x * 8 + wave;
  if (rt >= nRowTiles) return;

  int row  = rt * 16 + (lane & 15);
  int g2   = (lane >> 4) * 2;
  int ncol = lane & 15;

  v8f c0 = {}, c1 = {}, c2 = {}, c3 = {};
  const float* xrow = x + (size_t)row * 64;
  for (int kk = 0; kk < 64; kk += 4)
    wmma_k4(xrow, sW, kk, g2, ncol, c0, c1, c2, c3);

  int m0 = rt * 16 + 8 * (lane >> 4);
  for (int r = 0; r < 8; ++r) {
    float* o = out + (size_t)(m0 + r) * 64 + ncol;
    o[0]  = c0[r] + sb[ncol];
    o[16] = c1[r] + sb[ncol + 16];
    o[32] = c2[r] + sb[ncol + 32];
    o[48] = c3[r] + sb[ncol + 48];
  }
}

extern "C" void kernel_launch(void* const* d_in, const int* in_sizes, int n_in,
                              void* d_out, int out_size, void* d_ws, size_t ws_size,
                              hipStream_t stream) {
  const float* h      = (const float*)d_in[0];
  const int*   src    = (const int*)d_in[1];
  const int*   dst    = (const int*)d_in[2];
  const int*   fe     = (const int*)d_in[3];
  const int*   se     = (const int*)d_in[4];
  const float* Wb_src = (const float*)d_in[5];   // [2,64,64]
  const float* coef_s = (const float*)d_in[6];   // [2,2]
  const float* Wb_dst = (const float*)d_in[7];   // [2,64,64]
  const float* coef_d = (const float*)d_in[8];   // [2,2]
  const float* Wlin   = (const float*)d_in[9];   // [2,128,64]
  const float* blin   = (const float*)d_in[10];  // [2,64]
  const float* Wskip  = (const float*)d_in[11];  // [2,64,64]
  const float* bskip  = (const float*)d_in[12];  // [2,64]
  const float* Wout   = (const float*)d_in[13];  // [64,64]
  const float* bout   = (const float*)d_in[14];  // [64]
  float* out = (float*)d_out;

  const int N = 100000, E = 800000;
  const int rowTiles = N / 16;  // 6250 exact

  // Workspace layout (floats): ps[N*64] | pd[N*64] | n[N*128] | xbuf[N*64]
  float* ps   = (float*)d_ws;
  float* pd   = ps + (size_t)N * 64;
  float* nbuf = pd + (size_t)N * 64;
  float* xbuf = nbuf + (size_t)N * 128;
  float* zbuf = ps;  // z overwrites ps (ps/pd dead after edge scatter)

  dim3 blk(256);
  int gProj = (rowTiles * 2 + 7) / 8;          // 2 matrices, 8 waves/block
  int gTile = (rowTiles + 7) / 8;              // 1 wave per 16-row tile
  int gZero = (N * 128 + 255) / 256;
  int gEdge = (E * 64 + 255) / 256;            // 64 threads per edge
  int gNorm = (N + 7) / 8;                     // 1 wave per row

  const float* xcur = h;
  for (int l = 0; l < 2; ++l) {
    sagnn_proj_kernel<<<gProj, blk, 0, stream>>>(
        xcur, Wb_src + (size_t)l * 4096, Wb_dst + (size_t)l * 4096, ps, pd, rowTiles);
    sagnn_zero_kernel<<<gZero, blk, 0, stream>>>(nbuf, N * 128);
    sagnn_edge_kernel<<<gEdge, blk, 0, stream>>>(
        ps, pd, src, dst, fe, se, coef_s + l * 2, coef_d + l * 2, nbuf, E);
    sagnn_combine_kernel<<<gTile, blk, 0, stream>>>(
        nbuf, xcur, Wlin + (size_t)l * 8192, blin + (size_t)l * 64,
        Wskip + (size_t)l * 4096, bskip + (size_t)l * 64, zbuf, rowTiles);
    sagnn_normalize_kernel<<<gNorm, blk, 0, stream>>>(zbuf, xbuf, N);
    xcur = xbuf;
  }
  sagnn_out_kernel<<<gTile, blk, 0, stream>>>(xcur, Wout, bout, out, rowTiles);

  (void)in_sizes; (void)n_in; (void)out_size; (void)ws_size;
}